// LCNNConv2d_38457137169040
// MI455X (gfx1250) — compile-verified
//
#include <hip/hip_runtime.h>

// MI455X / gfx1250 wave32 WMMA implementation.
// out[b,o,hw] = sum_c W[o,c] * x[b,c,hw], with W = (sparse combine) @ dict.

typedef float v2f __attribute__((ext_vector_type(2)));
typedef float v8f __attribute__((ext_vector_type(8)));

#define C_IN   64
#define D_SIZE 512
#define O_OUT  256
#define HW     16384   // 128*128
#define BATCH  16
#define SPARS  4

// ---------------------------------------------------------------------------
// Phase 1: fold sparse lookup + dictionary into dense W[256][64].
// Gather formulation: W[o][c] = sum_s coeff[o,s] * dict[idx[o,s]][c].
// Duplicate indices sum naturally (== scatter-add coalescing in reference).
// ---------------------------------------------------------------------------
__global__ void __launch_bounds__(256) build_w_kernel(
        const float* __restrict__ dict,     // [512][64]
        const float* __restrict__ coeff,    // [256][4]
        const int*   __restrict__ idx,      // [256][4]
        float*       __restrict__ W)        // [256][64]
{
    const int t = blockIdx.x * 256 + threadIdx.x;   // 16384 threads
    const int o = t >> 6;
    const int c = t & 63;
    float acc = 0.0f;
#pragma unroll
    for (int s = 0; s < SPARS; ++s) {
        const int   d  = idx[o * SPARS + s];
        const float cf = coeff[o * SPARS + s];
        acc += cf * dict[d * C_IN + c];
    }
    W[o * C_IN + c] = acc;
}

// ---------------------------------------------------------------------------
// Phase 2: GEMM  out[b][256][16384] = W[256][64] * x[b][64][16384]
// One wave32 -> one 16(M) x 64(N) output tile; K=64 via 16 x WMMA 16x16x4 f32.
//
// Fragment layouts (CDNA5 ISA 7.12.2, 32-bit, wave32):
//   A 16x4 : VGPR v, lane l -> M = l&15, K = v + 2*(l>>4)
//   B  4x16: VGPR v, lane l -> K = v + 2*(l>>4), N = l&15
//   C 16x16: VGPR v, lane l -> M = v + 8*(l>>4), N = l&15
//
// Output is write-once (268 MB stream) -> non-temporal stores so L2 keeps
// x (16-way reuse across o-tiles) and W instead of the dead output stream.
// ---------------------------------------------------------------------------
__global__ void __launch_bounds__(256) lcnn_gemm_kernel(
        const float* __restrict__ x,    // [16][64][16384]
        const float* __restrict__ W,    // [256][64]
        float*       __restrict__ out)  // [16][256][16384]
{
    const int lane = threadIdx.x & 31;
    const int gw   = blockIdx.x * 8 + (threadIdx.x >> 5);   // 65536 waves

    // o_tile fastest so the 8 waves of a block share the same x columns (WGP$ reuse)
    const int o_tile  = gw & 15;           // 16 tiles of 16 output channels
    const int n_chunk = (gw >> 4) & 255;   // 256 chunks of 64 pixels
    const int b       = gw >> 12;          // 16 batch images

    const int half = lane >> 4;            // 0 or 1
    const int ln   = lane & 15;
    const int o0   = o_tile * 16;
    const int n0   = n_chunk * 64;

    // ---- A fragments: W tile rows o0..o0+15, all of K=0..63 (32 VGPRs) ----
    v2f a[16];
    {
        const float* wrow = W + (o0 + ln) * C_IN + 2 * half;
#pragma unroll
        for (int kb = 0; kb < 16; ++kb) {
            a[kb].x = wrow[4 * kb + 0];    // K = 4*kb + 2*half
            a[kb].y = wrow[4 * kb + 1];    // K = 4*kb + 2*half + 1
        }
    }

    v8f acc0 = {}, acc1 = {}, acc2 = {}, acc3 = {};

    const float* xb = x + (size_t)b * C_IN * HW;
#pragma unroll
    for (int kb = 0; kb < 16; ++kb) {
        const int    c  = 4 * kb + 2 * half;          // this lane's K rows
        const float* xp = xb + (size_t)c * HW + n0 + ln;
        v2f b0, b1, b2, b3;
        b0.x = xp[0];       b0.y = xp[HW];            // (K=c, K=c+1) x N sub-tile 0
        b1.x = xp[16];      b1.y = xp[HW + 16];
        b2.x = xp[32];      b2.y = xp[HW + 32];
        b3.x = xp[48];      b3.y = xp[HW + 48];

        acc0 = __builtin_amdgcn_wmma_f32_16x16x4_f32(false, a[kb], false, b0,
                                                     (short)0, acc0, false, false);
        acc1 = __builtin_amdgcn_wmma_f32_16x16x4_f32(false, a[kb], false, b1,
                                                     (short)0, acc1, false, false);
        acc2 = __builtin_amdgcn_wmma_f32_16x16x4_f32(false, a[kb], false, b2,
                                                     (short)0, acc2, false, false);
        acc3 = __builtin_amdgcn_wmma_f32_16x16x4_f32(false, a[kb], false, b3,
                                                     (short)0, acc3, false, false);
    }

    // ---- Store D tiles (non-temporal): row o0 + v + 8*half, col n0 + 16*j + ln
#pragma unroll
    for (int v = 0; v < 8; ++v) {
        float* orow = out + ((size_t)b * O_OUT + (o0 + v + 8 * half)) * (size_t)HW
                          + n0 + ln;
        __builtin_nontemporal_store(acc0[v], orow + 0);
        __builtin_nontemporal_store(acc1[v], orow + 16);
        __builtin_nontemporal_store(acc2[v], orow + 32);
        __builtin_nontemporal_store(acc3[v], orow + 48);
    }
}

// ---------------------------------------------------------------------------
extern "C" void kernel_launch(void* const* d_in, const int* in_sizes, int n_in,
                              void* d_out, int out_size, void* d_ws, size_t ws_size,
                              hipStream_t stream) {
    const float* x     = (const float*)d_in[0];   // [16,64,128,128]
    const float* dict  = (const float*)d_in[1];   // [512,64,1,1]
    const float* coeff = (const float*)d_in[2];   // [256,1,1,4]
    const int*   idx   = (const int*)d_in[3];     // [256,1,1,4] int32
    float* out = (float*)d_out;                   // [16,256,128,128]
    float* W   = (float*)d_ws;                    // 256*64 floats = 64 KB scratch

    build_w_kernel<<<64, 256, 0, stream>>>(dict, coeff, idx, W);
    lcnn_gemm_kernel<<<8192, 256, 0, stream>>>(x, W, out);
}